// MambaAbsoluteThresholdPruner_3582002725456
// MI455X (gfx1250) — compile-verified
//
#include <hip/hip_runtime.h>
#include <math.h>

typedef __attribute__((ext_vector_type(2))) float v2f;
typedef __attribute__((ext_vector_type(4))) float v4f;
typedef __attribute__((ext_vector_type(8))) float v8f;

constexpr int BB = 8;
constexpr int NN = 8192;
constexpr int CC = 512;

// ---------------------------------------------------------------------------
// Phase 1: imp[b,n] = || x[b,n,:] - prev_x[b,n,:] ||_2 via V_WMMA_F32_16X16X4_F32.
// One wave handles a 16-token tile; diag(acc) of acc += D * D^T (K-steps of 4)
// gives the squared norms. f32 A(16x4) and B(4x16) operands share the same
// physical lane layout when B = A^T, so the same register pair feeds both.
// b128 global loads (128-channel chunks); prev_x is streamed non-temporally
// (single use) while x stays temporal so the 192MB L2 can serve the phase-3
// re-read. LDS rows padded to 132 -> bank = (4r+k)%64, conflict-free.
// ---------------------------------------------------------------------------
__global__ __launch_bounds__(128) void imp_wmma_kernel(const float* __restrict__ x,
                                                       const float* __restrict__ px,
                                                       float* __restrict__ imp_out) {
  __shared__ float lds[4][16][132];           // 33792 bytes
  const int w      = threadIdx.x >> 5;        // wave in block (4 waves)
  const int lane   = threadIdx.x & 31;
  const int token0 = blockIdx.x * 64 + w * 16;
  const int row    = lane & 15;               // A-matrix M (and B-matrix N)
  const int khalf  = (lane >> 4) << 1;        // lanes 16-31 hold K+2,K+3

  v8f acc = {};

  for (int c0 = 0; c0 < CC; c0 += 128) {
    // coalesced fill: per row, the wave loads 128 contiguous channels (b128/lane)
#pragma unroll 8
    for (int r = 0; r < 16; ++r) {
      size_t base = (size_t)(token0 + r) * CC + c0 + (lane << 2);
      v4f xv = *reinterpret_cast<const v4f*>(x + base);
      v4f pv = __builtin_nontemporal_load(reinterpret_cast<const v4f*>(px + base));
      *reinterpret_cast<v4f*>(&lds[w][r][lane << 2]) = xv - pv;
    }
    __syncthreads();
    // 32 K-steps of 4 over this 128-channel chunk
#pragma unroll
    for (int t = 0; t < 32; ++t) {
      float2 af = *reinterpret_cast<const float2*>(&lds[w][row][t * 4 + khalf]);
      v2f a;
      a.x = af.x;
      a.y = af.y;
      acc = __builtin_amdgcn_wmma_f32_16x16x4_f32(
          /*neg_a=*/false, a, /*neg_b=*/false, a,
          /*c_mod=*/(short)0, acc, /*reuse_a=*/false, /*reuse_b=*/false);
    }
    __syncthreads();
  }

  // C/D layout: VGPR v -> lanes 0-15: (M=v, N=lane); lanes 16-31: (M=v+8, N=lane-16).
  // Dump to LDS, then pull the diagonal.
#pragma unroll
  for (int v = 0; v < 8; ++v) {
    int M = v + ((lane >> 4) << 3);
    lds[w][M][lane & 15] = acc[v];
  }
  __syncthreads();
  if (lane < 16) {
    imp_out[token0 + lane] = sqrtf(lds[w][lane][lane]);
  }
}

// ---------------------------------------------------------------------------
// Phase 2: per batch row -- threshold, keep mask, first-argmax fallback,
// stable ranks via inclusive scan, scatter src_of[rank] = token.
// One 1024-thread block per row; each thread owns 8 contiguous tokens.
// ---------------------------------------------------------------------------
__global__ __launch_bounds__(1024) void scan_kernel(const float* __restrict__ imp,
                                                    const float* __restrict__ kth,
                                                    const float* __restrict__ kbase,
                                                    float* __restrict__ thresh_out,
                                                    float* __restrict__ mask_out,
                                                    int* __restrict__ counts,
                                                    int* __restrict__ src_of) {
  const int b   = blockIdx.x;
  const int tid = threadIdx.x;
  const float threshold = fmaxf(kbase[0] + kth[0], 0.0f);
  if (b == 0 && tid == 0) thresh_out[0] = threshold;

  constexpr int PER = NN / 1024;              // 8 tokens per thread
  const int rowbase = b * NN;
  const int tok0    = tid * PER;

  int   k[PER];
  int   localsum = 0;
  float mx = -INFINITY;
  int   mi = 0;
#pragma unroll
  for (int i = 0; i < PER; ++i) {
    float v = imp[rowbase + tok0 + i];
    int kk  = (v >= threshold) ? 1 : 0;
    k[i] = kk;
    localsum += kk;
    if (v > mx) { mx = v; mi = tok0 + i; }    // strict > keeps first max in chunk
  }

  __shared__ int   ssum[1024];
  __shared__ float smax[1024];
  __shared__ int   sidx[1024];
  ssum[tid] = localsum;
  smax[tid] = mx;
  sidx[tid] = mi;
  __syncthreads();

  // first-argmax reduction (ties -> lower index, matching jnp.argmax)
  for (int s = 512; s > 0; s >>= 1) {
    if (tid < s) {
      float a = smax[tid], c = smax[tid + s];
      if (c > a || (c == a && sidx[tid + s] < sidx[tid])) {
        smax[tid] = c;
        sidx[tid] = sidx[tid + s];
      }
    }
    __syncthreads();
  }

  // Hillis-Steele inclusive scan of per-thread keep counts
  for (int off = 1; off < 1024; off <<= 1) {
    int add = (tid >= off) ? ssum[tid - off] : 0;
    __syncthreads();
    ssum[tid] += add;
    __syncthreads();
  }

  const int total = ssum[1023];
  const int excl  = ssum[tid] - localsum;
  const int maxi  = sidx[0];

  if (total == 0) {
    // fallback: keep only the first-argmax token
#pragma unroll
    for (int i = 0; i < PER; ++i) {
      int tok = tok0 + i;
      mask_out[rowbase + tok] = (tok == maxi) ? 1.0f : 0.0f;
    }
    if (tid == 0) {
      counts[b] = 1;
      src_of[rowbase] = maxi;
    }
  } else {
    int r = excl;
#pragma unroll
    for (int i = 0; i < PER; ++i) {
      int tok = tok0 + i;
      mask_out[rowbase + tok] = (float)k[i];
      if (k[i]) {
        src_of[rowbase + r] = tok;
        ++r;
      }
    }
    if (tid == 0) counts[b] = total;
  }
}

// ---------------------------------------------------------------------------
// Phase 3: stable compaction gather. One 128-thread block per output row;
// b128 per thread -> 512 floats. Rows >= count are zero-filled. x reads stay
// temporal (hoping for 192MB-L2 hits from phase 1); out stores are NT since
// the output is written exactly once and never re-read.
// ---------------------------------------------------------------------------
__global__ __launch_bounds__(128) void gather_kernel(const float* __restrict__ x,
                                                     const int* __restrict__ counts,
                                                     const int* __restrict__ src_of,
                                                     float* __restrict__ out) {
  const int blk = blockIdx.x;
  const int b   = blk >> 13;                  // / NN
  const int j   = blk & (NN - 1);
  const int cnt = counts[b];

  v4f val = {0.f, 0.f, 0.f, 0.f};
  if (j < cnt) {
    const int src = src_of[b * NN + j];
    const v4f* xp = reinterpret_cast<const v4f*>(x + (size_t)(b * NN + src) * CC);
    val = xp[threadIdx.x];
  }
  v4f* op = reinterpret_cast<v4f*>(out + (size_t)blk * CC);
  __builtin_nontemporal_store(val, op + threadIdx.x);
}

extern "C" void kernel_launch(void* const* d_in, const int* in_sizes, int n_in,
                              void* d_out, int out_size, void* d_ws, size_t ws_size,
                              hipStream_t stream) {
  const float* x     = (const float*)d_in[0];
  const float* px    = (const float*)d_in[1];
  const float* kth   = (const float*)d_in[2];   // keep_threshold (scalar)
  const float* kbase = (const float*)d_in[3];   // keep_threshold_base (scalar)

  // output layout: [out B*N*C][threshold 1][imp B*N][mask B*N]
  float* out        = (float*)d_out;
  float* thresh_out = out + (size_t)BB * NN * CC;
  float* imp_out    = thresh_out + 1;
  float* mask_out   = imp_out + (size_t)BB * NN;

  int* counts = (int*)d_ws;                     // BB ints
  int* src_of = counts + 16;                    // BB*NN ints (rank -> token)

  imp_wmma_kernel<<<(BB * NN) / 64, 128, 0, stream>>>(x, px, imp_out);
  scan_kernel<<<BB, 1024, 0, stream>>>(imp_out, kth, kbase, thresh_out,
                                       mask_out, counts, src_of);
  gather_kernel<<<BB * NN, 128, 0, stream>>>(x, counts, src_of, out);
}